// LocalTemporalCL_38019050504456
// MI455X (gfx1250) — compile-verified
//
#include <hip/hip_runtime.h>
#include <hip/hip_bf16.h>
#include <stdint.h>

// ---------------- problem constants ----------------
#define Bn   1024
#define Ln   512
#define Tn   256
#define Pn   8
#define TSn  8
#define HIDn 64
#define NOn  32
#define NFn  16          // NO/2
#define BN_EPS 1e-5f

typedef __attribute__((ext_vector_type(2))) float v2f;
typedef __attribute__((ext_vector_type(8))) float v8f;

struct TVals { int t[Pn]; };

// ---------------- K1: pooled means (bandwidth-bound part) ----------------
// one block per batch b; wave w == pool p (wave32 => uniform loop bound per wave)
__global__ void k_pool(const float* __restrict__ Main_x,
                       const float* __restrict__ Auxi_x,
                       float* __restrict__ mmean,     // [P][B]
                       float* __restrict__ enc,       // [TS][P][B]
                       TVals tv) {
    const int b    = blockIdx.x;
    const int tid  = threadIdx.x;       // 256 threads = 8 waves
    const int p    = tid >> 5;
    const int lane = tid & 31;
    const int tp   = tv.t[p];

    const float* base = Main_x + (size_t)b * (Ln * Tn) + p * 32 + lane;
    float s = 0.0f;
    int l = 0;
    // unrolled-by-4 strided accumulation over rows l = 0..tp
    for (; l + 3 <= tp; l += 4) {
        s += base[(size_t)(l + 0) * Tn];
        s += base[(size_t)(l + 1) * Tn];
        s += base[(size_t)(l + 2) * Tn];
        s += base[(size_t)(l + 3) * Tn];
    }
    for (; l <= tp; ++l) s += base[(size_t)l * Tn];
    #pragma unroll
    for (int m = 16; m > 0; m >>= 1) s += __shfl_xor(s, m, 32);
    if (lane == 0) mmean[p * Bn + b] = s / (32.0f * (float)(tp + 1));

    const float* abase = Auxi_x + (size_t)b * (Ln * Tn) + p * 32 + lane;
    #pragma unroll
    for (int sI = 0; sI < TSn; ++sI) {
        float e = abase[(size_t)(tp + sI + 1) * Tn];
        #pragma unroll
        for (int m = 16; m > 0; m >>= 1) e += __shfl_xor(e, m, 32);
        if (lane == 0) enc[sI * (Pn * Bn) + p * Bn + b] = e * (1.0f / 32.0f);
    }
}

// ---------------- K2: fold weights through the rank-1 embedding ----------------
// A[m] = <W_embed, W_map[m]>, C[m] = <b_embed, W_map[m]> + b_map[m]
// u[o] = <W_embed, W1[:,o]>,  v[o] = <b_embed, W1[:,o]> + b1[o]
__global__ void k_weights(const float* __restrict__ W_embed,
                          const float* __restrict__ b_embed,
                          const float* __restrict__ W_map,
                          const float* __restrict__ b_map,
                          const float* __restrict__ W1,
                          const float* __restrict__ b1,
                          float* __restrict__ A, float* __restrict__ C,
                          float* __restrict__ u, float* __restrict__ v) {
    const int tid = threadIdx.x;     // 64 threads
    if (tid < TSn) {
        float a = 0.0f, c = 0.0f;
        for (int h = 0; h < HIDn; ++h) {
            const float wm = W_map[tid * HIDn + h];
            a += W_embed[h] * wm;
            c += b_embed[h] * wm;
        }
        A[tid] = a;
        C[tid] = c + b_map[tid];
    }
    if (tid < NOn) {
        float uu = 0.0f, vv = 0.0f;
        for (int h = 0; h < HIDn; ++h) {
            const float w = W1[h * NOn + tid];
            uu += W_embed[h] * w;
            vv += b_embed[h] * w;
        }
        u[tid] = uu;
        v[tid] = vv + b1[tid];
    }
}

// ---------------- K3: per-pool BN stats of mmean -> fused gain g[p][o] ----------------
__global__ void k_stats(const float* __restrict__ mmean,
                        const float* __restrict__ u,
                        const float* __restrict__ gamma,
                        float* __restrict__ mbar,     // [P]
                        float* __restrict__ g) {      // [P][NO]
    __shared__ float s1[256], s2[256];
    __shared__ float vm_sh;
    const int p = blockIdx.x, tid = threadIdx.x;
    float a = 0.0f, b = 0.0f;
    for (int i = tid; i < Bn; i += 256) {
        const float x = mmean[p * Bn + i];
        a += x; b += x * x;
    }
    s1[tid] = a; s2[tid] = b;
    __syncthreads();
    for (int off = 128; off > 0; off >>= 1) {
        if (tid < off) { s1[tid] += s1[tid + off]; s2[tid] += s2[tid + off]; }
        __syncthreads();
    }
    if (tid == 0) {
        const float mb = s1[0] * (1.0f / (float)Bn);
        vm_sh = s2[0] * (1.0f / (float)Bn) - mb * mb;  // biased var (ddof=0)
        mbar[p] = mb;
    }
    __syncthreads();
    if (tid < NOn) {
        const float uo = u[tid];
        g[p * NOn + tid] = uo * rsqrtf(vm_sh * uo * uo + BN_EPS) * gamma[tid];
    }
}

// ---------------- K4: NCE logsumexp partial sums ----------------
// block = 4*(m*P+p) + chunk ; each thread handles one j, loops i over enc row in LDS
__global__ void k_nce(const float* __restrict__ enc,
                      const float* __restrict__ mmean,
                      const float* __restrict__ A,
                      const float* __restrict__ C,
                      float* __restrict__ bsum) {
    __shared__ float encs[Bn];
    __shared__ float redA[256];
    __shared__ float redB[256];
    const int blk   = blockIdx.x;
    const int pair  = blk >> 2;
    const int chunk = blk & 3;
    const int m = pair >> 3, p = pair & 7;
    const int tid = threadIdx.x;

    const float* er = enc + m * (Pn * Bn) + p * Bn;
    for (int i = tid; i < Bn; i += 256) encs[i] = er[i];
    __syncthreads();

    float mx = -3.0e38f, mn = 3.0e38f;
    for (int i = tid; i < Bn; i += 256) {
        mx = fmaxf(mx, encs[i]);
        mn = fminf(mn, encs[i]);
    }
    redA[tid] = mx; redB[tid] = mn;
    __syncthreads();
    for (int off = 128; off > 0; off >>= 1) {
        if (tid < off) {
            redA[tid] = fmaxf(redA[tid], redA[tid + off]);
            redB[tid] = fminf(redB[tid], redB[tid + off]);
        }
        __syncthreads();
    }
    mx = redA[0]; mn = redB[0];
    __syncthreads();

    const int   j     = chunk * 256 + tid;
    const float predj = mmean[p * Bn + j] * A[m] + C[m];
    const float M     = (predj >= 0.0f) ? mx * predj : mn * predj;  // max_i enc_i*pred_j
    float ssum = 0.0f;
    #pragma unroll 8
    for (int i = 0; i < Bn; ++i) ssum += __expf(encs[i] * predj - M);
    const float lse     = M + __logf(ssum);
    const float contrib = encs[j] * predj - lse;     // diag - lse

    redA[tid] = contrib;
    __syncthreads();
    for (int off = 128; off > 0; off >>= 1) {
        if (tid < off) redA[tid] += redA[tid + off];
        __syncthreads();
    }
    if (tid == 0) bsum[blk] = redA[0];
}

__global__ void k_final(const float* __restrict__ bsum, float* __restrict__ out0) {
    __shared__ float red[256];
    const int tid = threadIdx.x;
    red[tid] = bsum[tid];
    __syncthreads();
    for (int off = 128; off > 0; off >>= 1) {
        if (tid < off) red[tid] += red[tid + off];
        __syncthreads();
    }
    // mean_p( sum_{m,b}(diag-lse) / (-B*TS) ) = total / (-B*TS*P)
    if (tid == 0) out0[0] = red[0] * (-1.0f / (float)(Bn * TSn * Pn));
}

// ---------------- K5: FM = relu(BN(h)) @ W2 + b2 via V_WMMA_F32_16X16X4_F32 ----------------
// One wave32 per 16x16 output tile; K=NO=32 in 8 wmma steps; full f32 precision.
__global__ void k_fm(const float* __restrict__ mmean,
                     const float* __restrict__ mbar,
                     const float* __restrict__ g,
                     const float* __restrict__ beta,
                     const float* __restrict__ W2,
                     const float* __restrict__ b2,
                     float* __restrict__ out) {
    const int tid    = threadIdx.x;        // 128 = 4 waves
    const int wave   = tid >> 5;
    const int lane   = tid & 31;
    const int tile   = blockIdx.x * 4 + wave;   // 0..511 (8192 rows / 16)
    const int p      = tile >> 6;               // 64 tiles per pool
    const int hi     = lane >> 4;               // 0: lanes 0-15, 1: lanes 16-31
    const int lm     = lane & 15;
    const int row    = tile * 16 + lm;          // this lane's A-matrix M row

    const float cm = mmean[row] - mbar[p];      // centered mmean, rank-1 BN
    v8f acc = {0.f, 0.f, 0.f, 0.f, 0.f, 0.f, 0.f, 0.f};

    #pragma unroll
    for (int kk = 0; kk < 8; ++kk) {
        // A fragment: lanes 0-15 carry K = 4kk+{0,1}; lanes 16-31 carry K = 4kk+{2,3}
        const int o0 = kk * 4 + hi * 2;
        const float g0 = g[p * NOn + o0];
        const float g1 = g[p * NOn + o0 + 1];
        const float r0 = fmaxf(cm * g0 + beta[o0],     0.0f);
        const float r1 = fmaxf(cm * g1 + beta[o0 + 1], 0.0f);
        v2f a = {r0, r1};
        // B fragment (4x16): VGPR0 = row K(+0/+2), VGPR1 = row K(+1/+3), N = lm
        v2f b = {W2[o0 * NFn + lm], W2[(o0 + 1) * NFn + lm]};
        acc = __builtin_amdgcn_wmma_f32_16x16x4_f32(
            /*neg_a=*/false, a, /*neg_b=*/false, b,
            /*c_mod=*/(short)0, acc, /*reuse_a=*/false, /*reuse_b=*/false);
    }

    // D layout: VGPR r -> rows r (lanes 0-15) and r+8 (lanes 16-31), column = lm
    #pragma unroll
    for (int r = 0; r < 8; ++r) {
        const int orow = tile * 16 + r + hi * 8;
        out[1 + orow * NFn + lm] = acc[r] + b2[lm];
    }
}

// ---------------- host: threefry2x32 reproduction of jax.random.randint(key(42),(8,),0,504) ----------------
static inline uint32_t rotl32(uint32_t x, uint32_t n) { return (x << n) | (x >> (32u - n)); }

static void tf2x32(uint32_t k0, uint32_t k1, uint32_t c0, uint32_t c1,
                   uint32_t& o0, uint32_t& o1) {
    const uint32_t ks[3] = { k0, k1, k0 ^ k1 ^ 0x1BD11BDAu };
    static const uint32_t rotA[4] = {13, 15, 26, 6};
    static const uint32_t rotB[4] = {17, 29, 16, 24};
    uint32_t x0 = c0 + ks[0], x1 = c1 + ks[1];
    for (int d = 0; d < 5; ++d) {
        const uint32_t* R = (d & 1) ? rotB : rotA;
        for (int r = 0; r < 4; ++r) { x0 += x1; x1 = rotl32(x1, R[r]); x1 ^= x0; }
        x0 += ks[(d + 1) % 3];
        x1 += ks[(d + 2) % 3] + (uint32_t)(d + 1);
    }
    o0 = x0; o1 = x1;
}

extern "C" void kernel_launch(void* const* d_in, const int* in_sizes, int n_in,
                              void* d_out, int out_size, void* d_ws, size_t ws_size,
                              hipStream_t stream) {
    (void)in_sizes; (void)n_in; (void)out_size; (void)ws_size;

    const float* Main_x  = (const float*)d_in[0];
    const float* Auxi_x  = (const float*)d_in[1];
    const float* W_embed = (const float*)d_in[2];
    const float* b_embed = (const float*)d_in[3];
    const float* W_map   = (const float*)d_in[4];
    const float* b_map   = (const float*)d_in[5];
    const float* W1      = (const float*)d_in[6];
    const float* b1      = (const float*)d_in[7];
    const float* gamma   = (const float*)d_in[8];
    const float* beta    = (const float*)d_in[9];
    const float* W2      = (const float*)d_in[10];
    const float* b2      = (const float*)d_in[11];
    float*       out     = (float*)d_out;

    // --- t = jax.random.randint(jax.random.key(42), (8,), 0, 504) ---
    // key data = [0, 42]; split -> k1 (higher bits), k2 (lower bits);
    // randint: t = (hi*((65536%504)^2 % 504) + lo) mod 504 = (hi*256 + lo) mod 504
    uint32_t a0, a1, b0v, b1v;
    tf2x32(0u, 42u, 0u, 2u, a0, a1);   // split counters pair (0,2)
    tf2x32(0u, 42u, 1u, 3u, b0v, b1v); // split counters pair (1,3)
    const uint32_t k1_0 = a0,  k1_1 = b0v;   // first subkey
    const uint32_t k2_0 = a1,  k2_1 = b1v;   // second subkey
    uint32_t hi[8], lo[8];
    for (int i = 0; i < 4; ++i) {
        uint32_t o0, o1;
        tf2x32(k1_0, k1_1, (uint32_t)i, (uint32_t)(i + 4), o0, o1);
        hi[i] = o0; hi[i + 4] = o1;
        tf2x32(k2_0, k2_1, (uint32_t)i, (uint32_t)(i + 4), o0, o1);
        lo[i] = o0; lo[i + 4] = o1;
    }
    TVals tv;
    for (int i = 0; i < 8; ++i) tv.t[i] = (int)((hi[i] * 256u + lo[i]) % 504u);

    // --- workspace layout (floats) ---
    float* ws    = (float*)d_ws;
    float* mmean = ws;                    // 8192
    float* enc   = mmean + Pn * Bn;       // 65536
    float* A     = enc + TSn * Pn * Bn;   // 8
    float* C     = A + TSn;               // 8
    float* u     = C + TSn;               // 32
    float* v     = u + NOn;               // 32 (kept for layout clarity)
    float* mbar  = v + NOn;               // 8
    float* g     = mbar + Pn;             // 256
    float* bsum  = g + Pn * NOn;          // 256

    k_pool   <<<Bn, 256, 0, stream>>>(Main_x, Auxi_x, mmean, enc, tv);
    k_weights<<<1, 64, 0, stream>>>(W_embed, b_embed, W_map, b_map, W1, b1, A, C, u, v);
    k_stats  <<<Pn, 256, 0, stream>>>(mmean, u, gamma, mbar, g);
    k_nce    <<<256, 256, 0, stream>>>(enc, mmean, A, C, bsum);
    k_final  <<<1, 256, 0, stream>>>(bsum, out);
    k_fm     <<<128, 128, 0, stream>>>(mmean, mbar, g, beta, W2, b2, out);
}